// TextLevelGCN_32229434589775
// MI455X (gfx1250) — compile-verified
//
#include <hip/hip_runtime.h>

typedef __attribute__((ext_vector_type(2))) float v2f;
typedef __attribute__((ext_vector_type(4))) float v4f;
typedef __attribute__((ext_vector_type(8))) float v8f;

#define NGRAPHS 512
#define DIM     300            // feature dim (must match D from in_sizes; not checked)
#define DIM4    (DIM / 4)      // 75 float4 per row
#define ACC_WAVES 8            // waves per graph-accumulate block

// ---------------------------------------------------------------------------
// Kernel 1: histogram edges by graph id + per-node in-degree counts.
// ---------------------------------------------------------------------------
__global__ void hist_count_kernel(const int* __restrict__ dst,
                                  const int* __restrict__ graph_ids,
                                  float* __restrict__ cnt,
                                  int* __restrict__ hist, int E) {
    int e = blockIdx.x * blockDim.x + threadIdx.x;
    if (e < E) {
        int d = dst[e];
        atomicAdd(&cnt[d], 1.0f);
        atomicAdd(&hist[graph_ids[d]], 1);
    }
}

// ---------------------------------------------------------------------------
// Kernel 2: exclusive scan of 512-bin histogram (serial; negligible work).
// Also seeds the scatter cursors.
// ---------------------------------------------------------------------------
__global__ void scan_kernel(const int* __restrict__ hist,
                            int* __restrict__ offs,
                            int* __restrict__ cursor) {
    if (threadIdx.x == 0 && blockIdx.x == 0) {
        int run = 0;
        for (int i = 0; i < NGRAPHS; ++i) {
            offs[i]   = run;
            cursor[i] = run;
            run += hist[i];
        }
        offs[NGRAPHS] = run;
    }
}

// ---------------------------------------------------------------------------
// Kernel 3: permute edges into gid-grouped order; precompute per-edge
//   vidx = nodes_batch[src[e]]          (embedding row to gather)
//   coef = edge_embed[edges_batch[e]] / max(cnt[dst[e]], 1)
// so the accumulate kernel reads them as linear streams.
// ---------------------------------------------------------------------------
__global__ void order_kernel(const int* __restrict__ nodes_batch,
                             const int* __restrict__ edges_batch,
                             const int* __restrict__ src,
                             const int* __restrict__ dst,
                             const int* __restrict__ graph_ids,
                             const float* __restrict__ edge_embed,
                             const float* __restrict__ cnt,
                             int* __restrict__ cursor,
                             int* __restrict__ vidx,
                             float* __restrict__ coef, int E) {
    int e = blockIdx.x * blockDim.x + threadIdx.x;
    if (e < E) {
        int d   = dst[e];
        int gid = graph_ids[d];
        int pos = atomicAdd(&cursor[gid], 1);
        vidx[pos] = nodes_batch[src[e]];
        coef[pos] = edge_embed[edges_batch[e]] / fmaxf(cnt[d], 1.0f);
    }
}

// ---------------------------------------------------------------------------
// Kernel 4: per-graph accumulate. One block per graph; each of the 8 waves
// owns a private LDS slice (no atomics: lanes own disjoint dwords, plain
// ds b128 read-modify-write). Embedding rows gathered as coalesced b128
// (rows are 1200 B = 16 B aligned). Final cross-wave reduce -> g[gid,:],
// stored without atomics (block exclusively owns its gid).
// ---------------------------------------------------------------------------
__global__ void graph_accum_kernel(const int* __restrict__ offs,
                                   const int* __restrict__ vidx,
                                   const float* __restrict__ coef,
                                   const float* __restrict__ node_embed,
                                   float* __restrict__ g) {
    __shared__ __align__(16) float lg[ACC_WAVES * DIM];

    int gid  = blockIdx.x;
    int wid  = threadIdx.x >> 5;
    int lane = threadIdx.x & 31;

    // zero this block's LDS slices
    for (int i = threadIdx.x; i < ACC_WAVES * DIM; i += blockDim.x)
        lg[i] = 0.0f;
    __syncthreads();

    v4f* slice4 = (v4f*)&lg[wid * DIM];

    int beg = offs[gid], end = offs[gid + 1];
    for (int i = beg + wid; i < end; i += ACC_WAVES) {
        int   v = vidx[i];   // wave-uniform
        float s = coef[i];   // wave-uniform
        const v4f* __restrict__ row4 =
            (const v4f*)(node_embed + (long long)v * DIM);
        for (int f = lane; f < DIM4; f += 32) {
            v4f r = row4[f];
            v4f acc = slice4[f];
            acc.x += r.x * s;
            acc.y += r.y * s;
            acc.z += r.z * s;
            acc.w += r.w * s;
            slice4[f] = acc;
        }
    }
    __syncthreads();

    // reduce the 8 slices and store the graph row (non-atomic)
    for (int d = threadIdx.x; d < DIM; d += blockDim.x) {
        float t = 0.0f;
#pragma unroll
        for (int w = 0; w < ACC_WAVES; ++w)
            t += lg[w * DIM + d];
        g[(long long)gid * DIM + d] = t;
    }
}

// ---------------------------------------------------------------------------
// Kernel 5: out = relu(g) @ fc_w + fc_b   via V_WMMA_F32_16X16X4_F32.
// One wave per 16x16 output tile. grid = (M/16, ceil(C/16)), block = 32.
// A (16x4 f32): lane L holds A[M=L%16][K = 2*(L/16)+j] in component j.
// B (4x16 f32): lane L holds B[K = 2*(L/16)+j][N=L%16] in component j.
// D (16x16 f32): vgpr r of lane L holds D[M = r + 8*(L/16)][N = L%16].
// B loads are index-clamped + value-masked (no branch): EXEC stays all-1s.
// ---------------------------------------------------------------------------
__global__ void gemm_wmma_kernel(const float* __restrict__ g,
                                 const float* __restrict__ fc_w,
                                 const float* __restrict__ fc_b,
                                 float* __restrict__ out,
                                 int D, int C) {
    int lane = threadIdx.x;           // 0..31
    int m0   = blockIdx.x * 16;
    int n0   = blockIdx.y * 16;
    int half = lane >> 4;             // 0 or 1
    int l16  = lane & 15;

    int   n    = n0 + l16;
    int   nc   = (n < C) ? n : (C - 1);   // clamp index, mask value
    float mask = (n < C) ? 1.0f : 0.0f;

    const float* __restrict__ arow = g + (long long)(m0 + l16) * D;

    v8f acc = {};
    for (int k0 = 0; k0 < D; k0 += 4) {
        int ka = k0 + 2 * half;
        v2f a, b;
        a.x = fmaxf(arow[ka],     0.0f);           // fused ReLU
        a.y = fmaxf(arow[ka + 1], 0.0f);
        b.x = fc_w[ka * C + nc]       * mask;      // fc_w is [D,C] row-major
        b.y = fc_w[(ka + 1) * C + nc] * mask;
        acc = __builtin_amdgcn_wmma_f32_16x16x4_f32(
            false, a, false, b, (short)0, acc, false, false);
    }

    if (n < C) {
        float bias = fc_b[n];
#pragma unroll
        for (int r = 0; r < 8; ++r) {
            int m = m0 + r + 8 * half;
            out[(long long)m * C + n] = acc[r] + bias;
        }
    }
}

// ---------------------------------------------------------------------------
// Host-side launcher.
// Inputs (setup_inputs order):
//   0 nodes_batch [N] i32    1 edges_batch [E] i32   2 src [E] i32
//   3 dst [E] i32            4 graph_ids [N] i32     5 node_embed [V*D] f32
//   6 edge_embed [NE] f32    7 fc_w [D*C] f32        8 fc_b [C] f32
// Output: [NGRAPHS * C] f32
// Workspace layout (floats/ints, 4B units):
//   [ g : NGRAPHS*DIM ][ cnt : N ][ coef : E ][ vidx : E ]
//   [ hist : 512 ][ offs : 513 ][ cursor : 512 ]           ~5 MB total
// ---------------------------------------------------------------------------
extern "C" void kernel_launch(void* const* d_in, const int* in_sizes, int n_in,
                              void* d_out, int out_size, void* d_ws, size_t ws_size,
                              hipStream_t stream) {
    const int* nodes_batch = (const int*)d_in[0];
    const int* edges_batch = (const int*)d_in[1];
    const int* src         = (const int*)d_in[2];
    const int* dst         = (const int*)d_in[3];
    const int* graph_ids   = (const int*)d_in[4];
    const float* node_embed = (const float*)d_in[5];
    const float* edge_embed = (const float*)d_in[6];
    const float* fc_w       = (const float*)d_in[7];
    const float* fc_b       = (const float*)d_in[8];
    float* out = (float*)d_out;

    const int N = in_sizes[0];
    const int E = in_sizes[2];
    const int C = in_sizes[8];           // 20
    const int D = in_sizes[7] / C;       // 300 (== DIM)

    float* g      = (float*)d_ws;                       // NGRAPHS*DIM
    float* cnt    = g + (size_t)NGRAPHS * DIM;          // N
    float* coef   = cnt + N;                            // E
    int*   vidx   = (int*)(coef + E);                   // E
    int*   hist   = vidx + E;                           // 512
    int*   offs   = hist + NGRAPHS;                     // 513
    int*   cursor = offs + NGRAPHS + 1;                 // 512

    // zero cnt + hist (graph-capture-safe async memsets)
    hipMemsetAsync(cnt,  0, (size_t)N * sizeof(float), stream);
    hipMemsetAsync(hist, 0, (size_t)NGRAPHS * sizeof(int), stream);

    const int blk = 256;
    // 1) in-degree counts + gid histogram
    hist_count_kernel<<<(E + blk - 1) / blk, blk, 0, stream>>>(
        dst, graph_ids, cnt, hist, E);
    // 2) exclusive scan (512 bins) + cursor seed
    scan_kernel<<<1, 32, 0, stream>>>(hist, offs, cursor);
    // 3) permute edges into gid-grouped streams (vidx, coef)
    order_kernel<<<(E + blk - 1) / blk, blk, 0, stream>>>(
        nodes_batch, edges_batch, src, dst, graph_ids,
        edge_embed, cnt, cursor, vidx, coef, E);
    // 4) per-graph LDS accumulate (no global atomics on g)
    graph_accum_kernel<<<NGRAPHS, ACC_WAVES * 32, 0, stream>>>(
        offs, vidx, coef, node_embed, g);
    // 5) WMMA GEMM: relu(g) @ fc_w + fc_b
    {
        dim3 grd(NGRAPHS / 16, (C + 15) / 16);   // (32, 2)
        gemm_wmma_kernel<<<grd, 32, 0, stream>>>(g, fc_w, fc_b, out, D, C);
    }
}